// BaseLIDIA_86870008528957
// MI455X (gfx1250) — compile-verified
//
#include <hip/hip_runtime.h>
#include <cstdint>
#include <cstddef>

typedef float v2f __attribute__((ext_vector_type(2)));
typedef float v8f __attribute__((ext_vector_type(8)));

namespace {
constexpr int T_     = 2;
constexpr int PHW    = 536;             // patches_h == patches_w
constexpr int HWP    = PHW * PHW;       // 287296 patches per t
constexpr int PD     = 75;              // 3 * 5 * 5
constexpr int OHW    = 532;             // cropped output h/w
constexpr int NPIX   = OHW * OHW;       // 283024
constexpr int TILE   = 16;              // output tile per block
constexpr int PB     = TILE + 4;        // 20 staged patch rows/cols
constexpr int ROWF   = PB * PD;         // 1500 floats per staged patch-row
constexpr int NTILES = (OHW + TILE - 1) / TILE;  // 34
}

static __device__ __forceinline__ void wait_async0() {
#if __has_builtin(__builtin_amdgcn_s_wait_asynccnt)
  __builtin_amdgcn_s_wait_asynccnt(0);
#else
  asm volatile("s_wait_asynccnt 0" ::: "memory");
#endif
}

// ------------------------------------------------------------------
// Kernel 1: partial sums of noisy per (t,c): 64 chunks each, fully
// deterministic. Wave32 reduction done with V_WMMA_F32_16X16X4_F32:
// A(16x4) = lane values (VGPR0=v, VGPR1=0), B(4x16) = ones
//   => D[m,n] = v_m + v_{m+16}; sum 8 accum VGPRs + one shfl => total.
// ------------------------------------------------------------------
__global__ void __launch_bounds__(256) means_partial_kernel(
    const float* __restrict__ noisy, float* __restrict__ partial)
{
  const int b  = blockIdx.x;    // 0..383
  const int tc = b >> 6;        // 0..5  (t*3+c)
  const int k  = b & 63;        // chunk
  const float* src = noisy + (size_t)tc * NPIX;

  float s = 0.f;
  for (int i = k * 256 + threadIdx.x; i < NPIX; i += 64 * 256)
    s += src[i];

  // ones-matmul wave reduction (EXEC is all-ones here: no divergence)
  v2f a;  a.x = s;   a.y = 0.f;
  v2f bo; bo.x = 1.f; bo.y = 1.f;
  v8f c = {};
  c = __builtin_amdgcn_wmma_f32_16x16x4_f32(false, a, false, bo,
                                            (short)0, c, false, false);
  float ws = c[0] + c[1] + c[2] + c[3] + c[4] + c[5] + c[6] + c[7];
  ws += __shfl_down(ws, 16, 32);       // lane 0: full wave32 sum

  __shared__ float red[8];
  const int wid = threadIdx.x >> 5;
  if ((threadIdx.x & 31) == 0) red[wid] = ws;
  __syncthreads();
  if (threadIdx.x == 0) {
    float tot = 0.f;
#pragma unroll
    for (int w = 0; w < 8; ++w) tot += red[w];
    partial[b] = tot;
  }
}

// ------------------------------------------------------------------
// Kernel 2: fold. One block = 16x16 output tile for one t, 3 channels.
// Stage (20 x 20 patches) x 75 fp32 of deno + 400 pw floats into LDS
// with CDNA5 async global->LDS (b128/b32), then per-pixel gather.
// ------------------------------------------------------------------
__global__ void __launch_bounds__(256) fold_kernel(
    const float* __restrict__ deno, const float* __restrict__ pwts,
    const float* __restrict__ partial, float* __restrict__ out)
{
  extern __shared__ float sm[];
  float* sD = sm;                       // 30000 floats: [row r][patch cx][d]
  float* sW = sm + ROWF * PB;           // 400 floats:   [row r][patch cx]
  float* sM = sm + ROWF * PB + PB * PB; // 3 floats: channel means numerators

  const int tid = threadIdx.x;
  const int tx0 = blockIdx.x * TILE;
  const int ty0 = blockIdx.y * TILE;
  const int t   = blockIdx.z;

  if (tid < 3) sM[tid] = 0.f;
  __syncthreads();

  const unsigned long long denoT =
      (unsigned long long)(const void*)(deno + (size_t)t * HWP * PD);
  const unsigned long long pwT =
      (unsigned long long)(const void*)(pwts + (size_t)t * HWP);
  const unsigned ldsD = (unsigned)(uintptr_t)sD;   // low 32 bits = LDS offset
  const unsigned ldsW = (unsigned)(uintptr_t)sW;

  // Columns actually staged (edge tile x: 536-528 = 8 patches => 150 chunks)
  const int npatch = min(PB, PHW - tx0);
  const int nchunk = (npatch * PD) >> 2;           // 375 or 150 b128 chunks

  // ---- async stage deno: per row, contiguous 16B-aligned run ----
  for (int r = 0; r < PB; ++r) {
    const int py = min(ty0 + r, PHW - 1);
    const unsigned grow = (unsigned)(py * PHW + tx0) * 300u;   // bytes, 16B aligned
    const unsigned lrow = ldsD + (unsigned)r * (unsigned)(ROWF * 4);
    for (int q = tid; q < nchunk; q += 256) {
      const unsigned voff  = grow + ((unsigned)q << 4);
      const unsigned laddr = lrow + ((unsigned)q << 4);
      asm volatile("global_load_async_to_lds_b128 %0, %1, %2"
                   :: "v"(laddr), "v"(voff), "s"(denoT) : "memory");
    }
  }
  // ---- async stage patch weights (400 x b32) ----
  for (int q = tid; q < PB * PB; q += 256) {
    const int r  = q / PB, cx = q - r * PB;
    const int py = min(ty0 + r, PHW - 1);
    const int px = min(tx0 + cx, PHW - 1);
    const unsigned voff  = (unsigned)(py * PHW + px) << 2;
    const unsigned laddr = ldsW + ((unsigned)q << 2);
    asm volatile("global_load_async_to_lds_b32 %0, %1, %2"
                 :: "v"(laddr), "v"(voff), "s"(pwT) : "memory");
  }

  // ---- second-stage means reduction (overlaps with async transfers) ----
  if (tid < 192) {                       // 6 full waves: no partial-wave EXEC
    const int c = tid >> 6, k = tid & 63;
    float v = partial[(t * 3 + c) * 64 + k];
#pragma unroll
    for (int o = 16; o; o >>= 1) v += __shfl_down(v, o, 32);
    if ((tid & 31) == 0) atomicAdd(&sM[c], v);  // 2 commutative adds/channel
  }

  wait_async0();
  __syncthreads();

  // ---- gather / fold ----
  const int lx = tid & 15, ly = tid >> 4;
  const float msc = 2.f / (float)NPIX;
  const float m0 = sM[0] * msc - 1.f;
  const float m1 = sM[1] * msc - 1.f;
  const float m2 = sM[2] * msc - 1.f;

  float a0 = 0.f, a1 = 0.f, a2 = 0.f, cw = 0.f;
#pragma unroll
  for (int ii = 0; ii < 5; ++ii) {
    const float* rD = sD + (ly + ii) * ROWF;
    const float* rW = sW + (ly + ii) * PB;
#pragma unroll
    for (int jj = 0; jj < 5; ++jj) {
      const int p = lx + jj;
      const float w = rW[p];
      const float* dp = rD + p * PD + (24 - ii * 5 - jj);
      cw += w;
      a0 = fmaf(dp[0],  w, a0);
      a1 = fmaf(dp[25], w, a1);
      a2 = fmaf(dp[50], w, a2);
    }
  }

  const int ox = tx0 + lx, oy = ty0 + ly;
  if (ox < OHW && oy < OHW) {
    const int base = t * 3 * NPIX + oy * OHW + ox;
    out[base]            = (a0 / cw + m0) * 0.5f + 0.5f;
    out[base + NPIX]     = (a1 / cw + m1) * 0.5f + 0.5f;
    out[base + 2 * NPIX] = (a2 / cw + m2) * 0.5f + 0.5f;
  }
}

extern "C" void kernel_launch(void* const* d_in, const int* in_sizes, int n_in,
                              void* d_out, int out_size, void* d_ws, size_t ws_size,
                              hipStream_t stream) {
  (void)in_sizes; (void)n_in; (void)out_size; (void)ws_size;
  const float* noisy = (const float*)d_in[0];
  const float* deno  = (const float*)d_in[1];
  const float* pwts  = (const float*)d_in[2];
  // d_in[3] (inds) and the scalar dims are unused by the reference math.
  float* out     = (float*)d_out;
  float* partial = (float*)d_ws;        // 384 floats of scratch

  means_partial_kernel<<<dim3(6 * 64), dim3(256), 0, stream>>>(noisy, partial);

  const int ldsBytes = (ROWF * PB + PB * PB + 16) * 4;   // 121,664 B
  hipFuncSetAttribute((const void*)fold_kernel,
                      hipFuncAttributeMaxDynamicSharedMemorySize, ldsBytes);
  fold_kernel<<<dim3(NTILES, NTILES, T_), dim3(256), ldsBytes, stream>>>(
      deno, pwts, partial, out);
}